// OlmoModel_67800353734696
// MI455X (gfx1250) — compile-verified
//
#include <hip/hip_runtime.h>
#include <hip/hip_bf16.h>
#include <math.h>

// ---------------- model constants (match reference) ----------------
#define D      2048
#define H      16
#define HD     128
#define L      2
#define V      32000
#define BS     16
#define B      16
#define MAXCTX 1024
#define NBLK   (MAXCTX / BS)        // 64
#define FF     (4 * D)              // 8192
#define EPS    1e-5f
#define SCALE  0.08838834764831843f // 1/sqrt(128)

// ---------------- WMMA f32 16x16x4 ----------------
typedef __attribute__((ext_vector_type(2))) float v2f;
typedef __attribute__((ext_vector_type(8))) float v8f;

__device__ __forceinline__ v8f wmma_f32_4(v2f a, v2f b, v8f c) {
  // (neg_a, A, neg_b, B, c_mod, C, reuse_a, reuse_b)
  return __builtin_amdgcn_wmma_f32_16x16x4_f32(false, a, false, b, (short)0, c,
                                               false, false);
}

// ---------------- embedding gather ----------------
__global__ __launch_bounds__(256) void embed_kernel(const int* __restrict__ ids,
                                                    const float* __restrict__ wte,
                                                    float* __restrict__ x) {
  const int b = blockIdx.x;
  const float4* src = (const float4*)(wte + (size_t)ids[b] * D);
  float4* dst = (float4*)(x + (size_t)b * D);
  for (int i = threadIdx.x; i < D / 4; i += 256) dst[i] = src[i];
}

// ---------------- layernorm (no affine) ----------------
__global__ __launch_bounds__(256) void ln_kernel(const float* __restrict__ x,
                                                 float* __restrict__ t) {
  __shared__ float r1[256];
  __shared__ float r2[256];
  const int b = blockIdx.x, tid = threadIdx.x;
  const float* xr = x + (size_t)b * D;
  float s = 0.f, s2 = 0.f;
  for (int i = tid; i < D; i += 256) {
    float v = xr[i];
    s += v;
    s2 += v * v;
  }
  r1[tid] = s;
  r2[tid] = s2;
  __syncthreads();
  for (int off = 128; off > 0; off >>= 1) {
    if (tid < off) {
      r1[tid] += r1[tid + off];
      r2[tid] += r2[tid + off];
    }
    __syncthreads();
  }
  const float mu = r1[0] * (1.f / D);
  const float var = r2[0] * (1.f / D) - mu * mu;
  const float rstd = rsqrtf(var + EPS);
  float* tr = t + (size_t)b * D;
  for (int i = tid; i < D; i += 256) tr[i] = (xr[i] - mu) * rstd;
}

// ---------------- skinny GEMM: C[16,N] = A[16,K] @ W[K,N] (+resid) ----------
// K, N compile-time: all inner-loop loads get immediate offsets from one
// loop-carried pointer. 8 K-steps (K=32) staged per chunk for MLP depth.
template <int K, int N, bool RESID>
__global__ __launch_bounds__(256) void gemm16_nn(const float* __restrict__ A,
                                                 const float* __restrict__ W,
                                                 float* __restrict__ C,
                                                 const float* __restrict__ resid) {
  const int lane = threadIdx.x & 31;
  const int wave = threadIdx.x >> 5;
  const int n0 = (blockIdx.x * 8 + wave) * 16;
  const int m = lane & 15;           // row of A / C within tile
  const int n = lane & 15;           // col of B / C within tile
  const int koff = (lane >> 4) << 1; // 0 or 2

  const float* pA = A + (size_t)m * K + koff;        // +u*4 immediates
  const float* pB = W + (size_t)koff * N + n0 + n;   // +(u*4+{0,1})*N immediates

  v8f acc = {};
  constexpr int CH = 8; // K-steps per chunk (32 K-elems)
  for (int kb0 = 0; kb0 < K; kb0 += 4 * CH) {
    __builtin_prefetch(pB + (size_t)128 * N, 0, 0); // 128 K-elems ahead
    v2f a[CH];
    float bx[CH], by[CH];
#pragma unroll
    for (int u = 0; u < CH; ++u) {
      a[u] = *(const v2f*)(pA + u * 4);
      bx[u] = pB[(size_t)(u * 4) * N];
      by[u] = pB[(size_t)(u * 4 + 1) * N];
    }
#pragma unroll
    for (int u = 0; u < CH; ++u) {
      v2f b;
      b.x = bx[u];
      b.y = by[u];
      acc = wmma_f32_4(a[u], b, acc);
    }
    pA += 4 * CH;
    pB += (size_t)(4 * CH) * N;
  }

  const int mbase = (lane >> 4) << 3; // 0 or 8
  const float* pR = RESID ? resid + (size_t)mbase * N + n0 + n : nullptr;
  float* pC = C + (size_t)mbase * N + n0 + n;
#pragma unroll
  for (int j = 0; j < 8; ++j) {
    float v = acc[j];
    if (RESID) v += pR[(size_t)j * N];
    pC[(size_t)j * N] = v;
  }
}

// ---------------- logits: out[16,V] = T[16,D] @ wte[V,D]^T ------------------
// operand swap: A <- wte tile (16 vocab rows), B <- T^T; both frags are
// contiguous float2 loads; each lane's 8 outputs are contiguous -> 2x b128.
__global__ __launch_bounds__(256) void gemm16_logits(const float* __restrict__ T,
                                                     const float* __restrict__ wte,
                                                     float* __restrict__ out) {
  const int lane = threadIdx.x & 31;
  const int wave = threadIdx.x >> 5;
  const int v0 = (blockIdx.x * 8 + wave) * 16;
  const int m = lane & 15;           // vocab row within tile
  const int nb = lane & 15;          // batch column
  const int koff = (lane >> 4) << 1;

  const float* pA = wte + (size_t)(v0 + m) * D + koff;
  const float* pB = T + (size_t)nb * D + koff;

  v8f acc = {};
  constexpr int CH = 8;
  for (int kb0 = 0; kb0 < D; kb0 += 4 * CH) {
    __builtin_prefetch(pA + 128, 0, 0);
    v2f a[CH], b[CH];
#pragma unroll
    for (int u = 0; u < CH; ++u) {
      a[u] = *(const v2f*)(pA + u * 4);
      b[u] = *(const v2f*)(pB + u * 4);
    }
#pragma unroll
    for (int u = 0; u < CH; ++u) acc = wmma_f32_4(a[u], b[u], acc);
    pA += 4 * CH;
    pB += 4 * CH;
  }

  const int mbase = (lane >> 4) << 3; // 0 or 8; outputs contiguous in vocab
  float4 lo = make_float4(acc[0], acc[1], acc[2], acc[3]);
  float4 hi = make_float4(acc[4], acc[5], acc[6], acc[7]);
  float4* pO = (float4*)(out + (size_t)nb * V + v0 + mbase);
  pO[0] = lo;
  pO[1] = hi;
}

// ---------------- attention (paged KV, alibi, decode) -----------------------
// one block per (b,h), 128 threads. New token's K/V taken from kv_new
// (workspace) instead of mutating the cache input.
__global__ __launch_bounds__(128) void attn_kernel(
    const float* __restrict__ q,       // [B, D] (= B,H,HD)
    const float* __restrict__ kv_new,  // [B, 2*HD]  (k | v)
    const float* __restrict__ kcache,  // [NUMBLOCKS, BS, H, HD] (layer slice)
    const float* __restrict__ vcache,
    const int* __restrict__ block_tables, // [B, NBLK]
    const int* __restrict__ ctx_lens,     // [B]
    float* __restrict__ out) {            // [B, D]
  __shared__ __align__(16) float qs[HD];
  __shared__ float ps[MAXCTX];
  __shared__ float red[128];
  __shared__ int bt_s[NBLK];

  const int b = blockIdx.x >> 4;
  const int h = blockIdx.x & 15;
  const int tid = threadIdx.x;
  const int ctx = ctx_lens[b];

  if (tid < HD) qs[tid] = q[(size_t)b * D + h * HD + tid];
  if (tid < NBLK) bt_s[tid] = block_tables[b * NBLK + tid];
  __syncthreads();

  const float slope = exp2f(-(float)(h + 1) * 0.5f); // ALIBI_BIAS_MAX/H = 0.5

  // phase 1: scores
  for (int s = tid; s < ctx; s += 128) {
    const float* kp;
    if (s == ctx - 1) {
      kp = kv_new + (size_t)b * (2 * HD);
    } else {
      const int blk = bt_s[s >> 4];
      kp = kcache + ((((size_t)blk * BS + (s & 15)) * H) + h) * HD;
    }
    float acc = 0.f;
#pragma unroll 8
    for (int d0 = 0; d0 < HD; d0 += 4) {
      float4 kk = *(const float4*)(kp + d0);
      float4 qq = *(const float4*)(qs + d0);
      acc += qq.x * kk.x + qq.y * kk.y + qq.z * kk.z + qq.w * kk.w;
    }
    ps[s] = acc * SCALE + slope * (float)(s - (ctx - 1));
  }
  __syncthreads();

  // softmax: max
  float mx = -1e30f;
  for (int s = tid; s < ctx; s += 128) mx = fmaxf(mx, ps[s]);
  red[tid] = mx;
  __syncthreads();
  for (int off = 64; off > 0; off >>= 1) {
    if (tid < off) red[tid] = fmaxf(red[tid], red[tid + off]);
    __syncthreads();
  }
  mx = red[0];
  __syncthreads();

  // softmax: exp + sum
  float sum = 0.f;
  for (int s = tid; s < ctx; s += 128) {
    float e = __expf(ps[s] - mx);
    ps[s] = e;
    sum += e;
  }
  red[tid] = sum;
  __syncthreads();
  for (int off = 64; off > 0; off >>= 1) {
    if (tid < off) red[tid] += red[tid + off];
    __syncthreads();
  }
  const float inv = 1.f / red[0];
  __syncthreads();

  // phase 2: out[d] = sum_s p[s] * v[s][d]; threads = d (coalesced 512B rows).
  // Hoist page-table lookup per 16-token block; in-block rows are immediate
  // offsets (si * H*HD floats) from one base -> loads batch per block.
  const int d = tid;
  const int nfull = ctx - 1; // cached tokens; last token comes from kv_new
  float acc = 0.f;
  for (int s0 = 0; s0 < nfull; s0 += BS) {
    const int blk = bt_s[s0 >> 4];
    const float* base = vcache + ((size_t)blk * BS * H + h) * HD + d;
    const int lim = (nfull - s0 < BS) ? (nfull - s0) : BS;
    if (lim == BS) {
      float vv[BS];
#pragma unroll
      for (int si = 0; si < BS; ++si) vv[si] = base[(size_t)si * H * HD];
#pragma unroll
      for (int si = 0; si < BS; ++si) acc += ps[s0 + si] * vv[si];
    } else {
      for (int si = 0; si < lim; ++si)
        acc += ps[s0 + si] * base[(size_t)si * H * HD];
    }
  }
  acc += ps[nfull] * kv_new[(size_t)b * (2 * HD) + HD + d];
  out[(size_t)b * D + h * HD + d] = acc * inv;
}

// ---------------- silu gate: y = silu(h[:,FF/2:]) * h[:,:FF/2] --------------
__global__ __launch_bounds__(256) void silu_gate_kernel(const float* __restrict__ h,
                                                        float* __restrict__ y) {
  const int i = blockIdx.x * 256 + threadIdx.x; // i in [0, B*FF/2)
  const int b = i >> 12;                        // /4096
  const int j = i & 4095;
  const float a = h[(size_t)b * FF + j];
  const float g = h[(size_t)b * FF + (FF / 2) + j];
  const float s = g / (1.f + __expf(-g));
  y[i] = s * a;
}

// ---------------- host-side sequencing ----------------
extern "C" void kernel_launch(void* const* d_in, const int* in_sizes, int n_in,
                              void* d_out, int out_size, void* d_ws,
                              size_t ws_size, hipStream_t stream) {
  const int* input_ids = (const int*)d_in[0];
  // d_in[1] positions, d_in[4] slot_mapping: unused (derivable from ctx_lens)
  const int* block_tables = (const int*)d_in[2];
  const int* context_lens = (const int*)d_in[3];
  const float* wte  = (const float*)d_in[5];
  const float* Wq   = (const float*)d_in[6];
  const float* Wkv  = (const float*)d_in[7];
  const float* Wo   = (const float*)d_in[8];
  const float* Wff1 = (const float*)d_in[9];
  const float* Wff2 = (const float*)d_in[10];
  const float* kcaches = (const float*)d_in[11];
  const float* vcaches = (const float*)d_in[12];
  float* out = (float*)d_out;

  float* ws = (float*)d_ws;
  float* x  = ws;                  // 16*2048
  float* t  = x + B * D;           // 16*2048
  float* qb = t + B * D;           // 16*2048
  float* kv = qb + B * D;          // 16*256
  float* ao = kv + B * 2 * HD;     // 16*2048
  float* hf = ao + B * D;          // 16*8192
  float* yg = hf + B * FF;         // 16*4096

  const size_t cache_layer = (size_t)(B * NBLK) * BS * H * HD; // elems/layer

  embed_kernel<<<B, 256, 0, stream>>>(input_ids, wte, x);

  for (int l = 0; l < L; ++l) {
    ln_kernel<<<B, 256, 0, stream>>>(x, t);
    // q = t @ Wq[l]           (16 x 2048 x 2048)
    gemm16_nn<D, D, false><<<D / 128, 256, 0, stream>>>(
        t, Wq + (size_t)l * D * D, qb, nullptr);
    // kv = t @ Wkv[l]         (16 x 2048 x 256)
    gemm16_nn<D, 2 * HD, false><<<(2 * HD) / 128, 256, 0, stream>>>(
        t, Wkv + (size_t)l * D * 2 * HD, kv, nullptr);
    // paged attention with alibi; new K/V read from `kv`
    attn_kernel<<<B * H, 128, 0, stream>>>(qb, kv, kcaches + l * cache_layer,
                                           vcaches + l * cache_layer,
                                           block_tables, context_lens, ao);
    // x += ao @ Wo[l]
    gemm16_nn<D, D, true><<<D / 128, 256, 0, stream>>>(
        ao, Wo + (size_t)l * D * D, x, x);
    // h = ln(x) @ Wff1[l]     (16 x 2048 x 8192)
    ln_kernel<<<B, 256, 0, stream>>>(x, t);
    gemm16_nn<D, FF, false><<<FF / 128, 256, 0, stream>>>(
        t, Wff1 + (size_t)l * D * FF, hf, nullptr);
    // y = silu(g) * a
    silu_gate_kernel<<<(B * FF / 2) / 256, 256, 0, stream>>>(hf, yg);
    // x += y @ Wff2[l]        (16 x 4096 x 2048)
    gemm16_nn<FF / 2, D, true><<<D / 128, 256, 0, stream>>>(
        yg, Wff2 + (size_t)l * (FF / 2) * D, x, x);
  }

  // logits = ln(x) @ wte^T    (16 x 2048 x 32000)
  ln_kernel<<<B, 256, 0, stream>>>(x, t);
  gemm16_logits<<<V / 128, 256, 0, stream>>>(t, wte, out);
}